// AblationScorer_77962246357091
// MI455X (gfx1250) — compile-verified
//
#include <hip/hip_runtime.h>
#include <stdint.h>

// ---------------------------------------------------------------------------
// AblationScorer: out[b,e,0] = 0.0f if segment e of row b is in the top-512
// of jax.random.uniform(key(42),(4096,1024)) for that row, else -inf.
// Input x is never read (reference uses only its shape) -> kernel is bound by
// the 16 MiB output write stream (~0.7 us @ 23.3 TB/s).
// ---------------------------------------------------------------------------

#define B_DIM  4096
#define E_DIM  1024
#define K_KEEP 512
#define HALF_N 2097152u            // (B_DIM * E_DIM) / 2

// jax.random.key(42) -> threefry key (0, 42)
#define TF_K0 0u
#define TF_K1 42u
#define TF_K2 (TF_K0 ^ TF_K1 ^ 0x1BD11BDAu)

__device__ __forceinline__ uint32_t rotl32(uint32_t x, uint32_t r) {
  return (x << r) | (x >> (32u - r));
}

// Exact JAX threefry2x32 for flat output index n of a size-(B*E) iota counter.
// counts split in halves: pair p = (x0=p, x1=p+HALF); out = concat(r0, r1).
__device__ __forceinline__ uint32_t threefry_bits(uint32_t n) {
  const uint32_t p  = (n < HALF_N) ? n : (n - HALF_N);
  uint32_t x0 = p + TF_K0;
  uint32_t x1 = (p + HALF_N) + TF_K1;
#define TF_ROUND(r) { x0 += x1; x1 = rotl32(x1, (r)); x1 ^= x0; }
  TF_ROUND(13) TF_ROUND(15) TF_ROUND(26) TF_ROUND(6)
  x0 += TF_K1; x1 += TF_K2 + 1u;
  TF_ROUND(17) TF_ROUND(29) TF_ROUND(16) TF_ROUND(24)
  x0 += TF_K2; x1 += TF_K0 + 2u;
  TF_ROUND(13) TF_ROUND(15) TF_ROUND(26) TF_ROUND(6)
  x0 += TF_K0; x1 += TF_K1 + 3u;
  TF_ROUND(17) TF_ROUND(29) TF_ROUND(16) TF_ROUND(24)
  x0 += TF_K1; x1 += TF_K2 + 4u;
  TF_ROUND(13) TF_ROUND(15) TF_ROUND(26) TF_ROUND(6)
  x0 += TF_K2; x1 += TF_K0 + 5u;
#undef TF_ROUND
  return (n < HALF_N) ? x0 : x1;
}

__global__ __launch_bounds__(E_DIM) void AblationScorer_77962246357091_kernel(
    float* __restrict__ out) {
  const uint32_t b = blockIdx.x;      // row
  const uint32_t t = threadIdx.x;     // segment within row
  const uint32_t n = b * E_DIM + t;   // flat PRNG index

  __shared__ uint32_t key23[E_DIM];   // 23-bit comparison keys
  __shared__ uint32_t hist[256];      // histogram of top-8 bits
  __shared__ uint32_t s_tstar, s_base;
  __shared__ float    outv[E_DIM];    // staged output tile (for async store)

  if (t < 256) hist[t] = 0u;
  __syncthreads();

  // uniform(bits) = bitcast((bits>>9)|0x3f800000) - 1 : strictly monotone in
  // key = bits>>9, ties in float <=> ties in key. Compare integer keys.
  const uint32_t key = threefry_bits(n) >> 9;
  key23[t] = key;
  atomicAdd(&hist[key >> 15], 1u);
  __syncthreads();

  // Serial 256-step suffix count (one lane): find threshold bin t* such that
  // base = #elements in bins > t*  <  K_KEEP  <=  base + hist[t*].
  if (t == 0) {
    uint32_t acc = 0, ts = 0;
    for (int bin = 255; bin >= 0; --bin) {
      const uint32_t c = hist[bin];
      if (acc + c >= K_KEEP) { ts = (uint32_t)bin; break; }
      acc += c;
    }
    s_tstar = ts;
    s_base  = acc;
  }
  __syncthreads();

  const uint32_t tstar = s_tstar;
  const uint32_t base  = s_base;
  const uint32_t mybin = key >> 15;

  bool keep;
  if (mybin != tstar) {
    keep = (mybin > tstar);
  } else {
    // Tie-aware rank inside the threshold bin only (few candidates):
    // count bin-mates that beat me (greater key, or equal key with lower
    // index) -- matches lax.top_k's stable, lower-index-first tie handling.
    uint32_t r = 0;
    for (uint32_t j = 0; j < E_DIM; ++j) {
      const uint32_t kj = key23[j];
      if ((kj >> 15) == tstar && (kj > key || (kj == key && j < t))) ++r;
    }
    keep = (base + r) < K_KEEP;
  }

  const float val = keep ? 0.0f : -__builtin_inff();

#if __has_builtin(__builtin_amdgcn_global_store_async_from_lds_b32)
  // CDNA5 async path: stage tile in LDS, stream LDS->global via the async
  // store unit (ASYNCcnt). Completion is waited for with s_wait_asynccnt(0)
  // when that builtin exists.
  outv[t] = val;
  __syncthreads();  // ensure ds_store retired before async engine reads LDS
  {
    typedef __attribute__((address_space(1))) int* gptr_t;
    typedef __attribute__((address_space(3))) int* lptr_t;
    gptr_t g = (gptr_t)(uint64_t)(out + n);
    lptr_t l = (lptr_t)(uint32_t)(uint64_t)(&outv[t]);
    __builtin_amdgcn_global_store_async_from_lds_b32(g, l, 0, 0);
  }
#if __has_builtin(__builtin_amdgcn_s_wait_asynccnt)
  __builtin_amdgcn_s_wait_asynccnt(0);
#endif
#else
  out[n] = val;
  (void)outv;
#endif
}

extern "C" void kernel_launch(void* const* d_in, const int* in_sizes, int n_in,
                              void* d_out, int out_size, void* d_ws, size_t ws_size,
                              hipStream_t stream) {
  (void)d_in; (void)in_sizes; (void)n_in; (void)d_ws; (void)ws_size; (void)out_size;
  float* out = (float*)d_out;  // 4096*1024 floats
  hipLaunchKernelGGL(AblationScorer_77962246357091_kernel,
                     dim3(B_DIM), dim3(E_DIM), 0, stream, out);
}